// Ordinal_Layer_12850542149872
// MI455X (gfx1250) — compile-verified
//
#include <hip/hip_runtime.h>

// Problem constants (fixed by the reference)
#define N      4096        // W*H = 64*64
#define NT     40          // number of thresholds
#define QPAD   64          // thresholds padded to power-of-two with +inf
#define NINV   41          // inverse-quantization LUT entries
#define ROWS   4           // output rows (a values) per block
#define TPB    256         // 8 waves of 32

// Native clang vector types (HIP's float4 is a struct and is rejected by
// __builtin_nontemporal_store; ext_vector_type works and has .x/.y/.z/.w).
typedef float v4f __attribute__((ext_vector_type(4)));

// Vector type matching the async-LDS builtin's parameter type.
typedef int v4i_vs __attribute__((vector_size(4 * sizeof(int))));
typedef __attribute__((address_space(1))) v4i_vs* gptr_v4i;
typedef __attribute__((address_space(3))) v4i_vs* lptr_v4i;

// Branchless rank-in-sorted-array + LUT gather.
// lq : replicated padded thresholds, lq[idx*32 + lane]  (idx in [0,64))
// li : replicated inv LUT,           li[idx*32 + lane]  (idx in [0,41))
// Lane-strided replication => lane l only ever touches banks {l, l+32}:
// conflict-free per wave regardless of per-lane divergent idx.
__device__ __forceinline__ float lut_lookup(float x, int lane,
                                            const float* __restrict__ lq,
                                            const float* __restrict__ li) {
    int pos = 0;
#pragma unroll
    for (int s = 32; s >= 1; s >>= 1) {
        float qv = lq[((pos + s - 1) << 5) + lane];
        pos += (x >= qv) ? s : 0;   // counts q[i] <= x; +inf padding never counts
    }
    return li[(pos << 5) + lane];
}

// Correctly-rounded f32 quotient db/da from a correctly-rounded reciprocal
// y = RN(1/da) via one Markstein fma refinement (exact for normal operands).
__device__ __forceinline__ float exact_div(float db, float da, float y) {
    float r0 = db * y;                      // within 1 ulp
    float e  = __builtin_fmaf(-da, r0, db); // exact remainder (single rounding)
    return __builtin_fmaf(e, y, r0);        // = RN(db/da)
}

__global__ __launch_bounds__(TPB) void Ordinal_Layer_kernel(
    const float* __restrict__ d,    // 4096 depths
    const float* __restrict__ q,    // 40 ascending thresholds
    const float* __restrict__ inv,  // 41-entry LUT
    float* __restrict__ out) {      // 4096*4096 f32
    __shared__ __align__(16) float lds_d[N];            // 16 KB staged depth map
    __shared__ __align__(16) float lds_q[QPAD * 32];    //  8 KB replicated thresholds
    __shared__ __align__(16) float lds_inv[NINV * 32];  // ~5.25 KB replicated LUT

    const int tid  = threadIdx.x;
    const int lane = tid & 31;

    // Build replicated, +inf-padded threshold table and replicated inv LUT.
    for (int t = tid; t < QPAD * 32; t += TPB) {
        const int idx = t >> 5;
        lds_q[t] = (idx < NT) ? q[idx] : 3.402823466e+38f;
    }
    for (int t = tid; t < NINV * 32; t += TPB) {
        lds_inv[t] = inv[t >> 5];
    }

    // Stage the 16 KB depth map into LDS.
#if defined(__gfx1250__) && __has_builtin(__builtin_amdgcn_global_load_async_to_lds_b128)
    // CDNA5 async global->LDS DMA path (ASYNCcnt-tracked, no VGPR data movement).
#pragma unroll
    for (int k = 0; k < N; k += TPB * 4) {
        const int idx = k + tid * 4;
        __builtin_amdgcn_global_load_async_to_lds_b128(
            (gptr_v4i)(d + idx),      // C-style cast: drops const, retypes to int4 AS1*
            (lptr_v4i)(lds_d + idx),  // generic->AS3 + retype to int4 AS3*
            0, 0);
    }
#if __has_builtin(__builtin_amdgcn_s_wait_asynccnt)
    __builtin_amdgcn_s_wait_asynccnt(0);
#else
    asm volatile("s_wait_asynccnt 0" ::: "memory");
#endif
#else
    // Fallback: plain cooperative copy.
    for (int t = tid; t < N; t += TPB) lds_d[t] = d[t];
#endif
    __syncthreads();

    const int a0 = blockIdx.x * ROWS;
#pragma unroll
    for (int r = 0; r < ROWS; ++r) {
        const int a = a0 + r;
        const float da   = lds_d[a];
        const float inva = 1.0f / da;   // one precise (correctly rounded) recip per row
        float* __restrict__ orow = out + (size_t)a * N;

        // 4096 outputs per row; 256 threads x float4 => 4 passes.
#pragma unroll
        for (int p = 0; p < N / (TPB * 4); ++p) {
            const int b0 = p * (TPB * 4) + tid * 4;
            const v4f db = *(const v4f*)(lds_d + b0);   // ds_load_b128, conflict-free
            v4f o;
            o.x = lut_lookup(exact_div(db.x, da, inva), lane, lds_q, lds_inv);
            o.y = lut_lookup(exact_div(db.y, da, inva), lane, lds_q, lds_inv);
            o.z = lut_lookup(exact_div(db.z, da, inva), lane, lds_q, lds_inv);
            o.w = lut_lookup(exact_div(db.w, da, inva), lane, lds_q, lds_inv);
            // Streaming 64 MB output: non-temporal coalesced b128 stores.
            __builtin_nontemporal_store(o, (v4f*)(orow + b0));
        }
    }
}

extern "C" void kernel_launch(void* const* d_in, const int* in_sizes, int n_in,
                              void* d_out, int out_size, void* d_ws, size_t ws_size,
                              hipStream_t stream) {
    (void)in_sizes; (void)n_in; (void)d_ws; (void)ws_size; (void)out_size;
    const float* d   = (const float*)d_in[0];   // (1,1,64,64) depth map, flat 4096
    const float* q   = (const float*)d_in[1];   // 40 thresholds
    const float* inv = (const float*)d_in[2];   // 41-entry LUT
    float* out = (float*)d_out;                 // 64*64*4096 f32

    dim3 grid(N / ROWS);   // 1024 blocks
    dim3 block(TPB);       // 256 threads = 8 wave32
    Ordinal_Layer_kernel<<<grid, block, 0, stream>>>(d, q, inv, out);
}